// IWD_ProjLayer_65876208386401
// MI455X (gfx1250) — compile-verified
//
#include <hip/hip_runtime.h>
#include <cstdint>

// IWD (inverse-distance-weighted) gather-interpolation for MI455X (gfx1250).
//
// out[b,n,c] = sum_k w[n,k] * x[b, nn_idx[n,k], c],  w = (1/(d+eps)) / sum
// B=4, N_in=49152, C=64, N_out=196608, K=4.
//
// Roofline: 192 MiB out + 48 MiB x + 6 MiB idx/dist ~= 11 us at 23.3 TB/s.
// x (48 MiB) fits in the 192 MB L2 -> the random K=4 gather is L2-resident
// after first touch. This kernel is pure data movement:
//   * 16 lanes x b128 cover one 64-float row: every gather is one coalesced
//     global_load_b128, every output write one global_store_b128.
//   * non-temporal stores for the 192 MiB output stream so x is not evicted
//     from L2 by the streaming writes.
//   * per-tile nn_idx/nn_dist staged once via CDNA5 async global->LDS copies
//     (global_load_async_to_lds_b128 + s_wait_asynccnt).
//   * weights normalized ONCE per point by wave 0 (no block barrier needed
//     before that: the same wave issued the async copies and waited on
//     ASYNCcnt), published through LDS, read back as one ds_load_b128.
//   * indices/weights reused across the unrolled batch loop -> 16 independent
//     b128 gathers in flight per thread.

typedef float f4 __attribute__((ext_vector_type(4)));
typedef int   i4 __attribute__((ext_vector_type(4)));

#define IWD_EPS 1e-8f

template <int BATCH>
__global__ __launch_bounds__(256)
void iwd_gather_kernel(const float* __restrict__ x,
                       const int*   __restrict__ nn_idx,
                       const float* __restrict__ nn_dist,
                       float*       __restrict__ out,
                       int n_in, int n_out)
{
    constexpr int C = 64;   // channels per point (one 256B row)
    constexpr int K = 4;    // neighbors

    __shared__ alignas(16) int   s_idx [16 * K];
    __shared__ alignas(16) float s_dist[16 * K];
    __shared__ alignas(16) float s_w   [16 * K];  // normalized weights

    const int tid = threadIdx.x;
    const int n0  = blockIdx.x << 4;   // 16 output points per block

    // ---- wave 0: async-stage idx/dist tile, then normalize weights -------
    if (tid < 32) {
        unsigned    lds_addr;
        const void* gaddr;
        if (tid < 16) {
            int n = n0 + tid; if (n >= n_out) n = n_out - 1;
            gaddr    = (const void*)(nn_idx + (size_t)n * K);
            lds_addr = (unsigned)(uintptr_t)&s_idx[tid * K];
        } else {
            int n = n0 + (tid - 16); if (n >= n_out) n = n_out - 1;
            gaddr    = (const void*)(nn_dist + (size_t)n * K);
            lds_addr = (unsigned)(uintptr_t)&s_dist[(tid - 16) * K];
        }
        // VDST = VGPR with LDS byte address, VADDR = 64-bit global address,
        // saddr disabled ("off"). Tracked by ASYNCcnt.
        asm volatile("global_load_async_to_lds_b128 %0, %1, off"
                     :: "v"(lds_addr), "v"(gaddr) : "memory");
        asm volatile("s_wait_asynccnt 0x0" ::: "memory");
        // Same wave issued + waited -> its LDS view is coherent already.
        if (tid < 16) {
            const float v0 = 1.0f / (s_dist[tid * K + 0] + IWD_EPS);
            const float v1 = 1.0f / (s_dist[tid * K + 1] + IWD_EPS);
            const float v2 = 1.0f / (s_dist[tid * K + 2] + IWD_EPS);
            const float v3 = 1.0f / (s_dist[tid * K + 3] + IWD_EPS);
            const float s  = 1.0f / (v0 + v1 + v2 + v3);
            s_w[tid * K + 0] = v0 * s;
            s_w[tid * K + 1] = v1 * s;
            s_w[tid * K + 2] = v2 * s;
            s_w[tid * K + 3] = v3 * s;
        }
    }
    __syncthreads();

    const int rs = tid >> 4;   // which of the 16 output points
    const int cg = tid & 15;   // 4-float channel group within the row
    const int n  = n0 + rs;
    if (n >= n_out) return;

    // One ds_load_b128 each (16-lane broadcast of the same 16B).
    const i4 iv = *(const i4*)&s_idx[rs * K];
    const f4 wv = *(const f4*)&s_w[rs * K];

    // 32-bit byte offsets within one batch slice of x (max ~12.6 MB).
    const unsigned coffB = (unsigned)cg * 16u;            // 16B per lane
    const unsigned r0 = (unsigned)iv.x * (C * 4u) + coffB;
    const unsigned r1 = (unsigned)iv.y * (C * 4u) + coffB;
    const unsigned r2 = (unsigned)iv.z * (C * 4u) + coffB;
    const unsigned r3 = (unsigned)iv.w * (C * 4u) + coffB;

    const size_t bstrideB = (size_t)n_in  * C * 4;        // x batch stride
    const size_t ostrideB = (size_t)n_out * C * 4;        // out batch stride
    const char*  xb   = (const char*)x;
    size_t       ob   = (size_t)n * (C * 4) + coffB;      // out byte offset

#pragma unroll
    for (int b = 0; b < BATCH; ++b) {
        const f4 g0 = *(const f4*)(xb + r0);              // b128 gather (L2)
        const f4 g1 = *(const f4*)(xb + r1);
        const f4 g2 = *(const f4*)(xb + r2);
        const f4 g3 = *(const f4*)(xb + r3);
        const f4 r  = g0 * wv.x + g1 * wv.y + g2 * wv.z + g3 * wv.w;
        // Streaming result: NT store keeps x resident in L2.
        __builtin_nontemporal_store(r, (f4*)((char*)out + ob));
        xb += bstrideB;
        ob += ostrideB;
    }
}

// Runtime-batch fallback (reference always has B=4; keep a generic path).
__global__ __launch_bounds__(256)
void iwd_gather_kernel_rt(const float* __restrict__ x,
                          const int*   __restrict__ nn_idx,
                          const float* __restrict__ nn_dist,
                          float*       __restrict__ out,
                          int n_in, int n_out, int batch)
{
    constexpr int C = 64, K = 4;
    const int tid = threadIdx.x;
    const int n   = (blockIdx.x << 4) + (tid >> 4);
    const int cg  = tid & 15;
    if (n >= n_out) return;

    const int   i0 = nn_idx[(size_t)n * K + 0];
    const int   i1 = nn_idx[(size_t)n * K + 1];
    const int   i2 = nn_idx[(size_t)n * K + 2];
    const int   i3 = nn_idx[(size_t)n * K + 3];
    const float v0 = 1.0f / (nn_dist[(size_t)n * K + 0] + IWD_EPS);
    const float v1 = 1.0f / (nn_dist[(size_t)n * K + 1] + IWD_EPS);
    const float v2 = 1.0f / (nn_dist[(size_t)n * K + 2] + IWD_EPS);
    const float v3 = 1.0f / (nn_dist[(size_t)n * K + 3] + IWD_EPS);
    const float s  = 1.0f / (v0 + v1 + v2 + v3);
    const float w0 = v0 * s, w1 = v1 * s, w2 = v2 * s, w3 = v3 * s;

    const size_t coff = (size_t)cg * 4;
    const size_t r0 = (size_t)i0 * C + coff, r1 = (size_t)i1 * C + coff;
    const size_t r2 = (size_t)i2 * C + coff, r3 = (size_t)i3 * C + coff;
    const size_t bstride = (size_t)n_in  * C;
    const size_t ostride = (size_t)n_out * C;
    const size_t obase   = (size_t)n * C + coff;

    for (int b = 0; b < batch; ++b) {
        const float* xb = x + (size_t)b * bstride;
        const f4 g0 = *(const f4*)(xb + r0);
        const f4 g1 = *(const f4*)(xb + r1);
        const f4 g2 = *(const f4*)(xb + r2);
        const f4 g3 = *(const f4*)(xb + r3);
        const f4 r  = g0 * w0 + g1 * w1 + g2 * w2 + g3 * w3;
        __builtin_nontemporal_store(r, (f4*)(out + (size_t)b * ostride + obase));
    }
}

extern "C" void kernel_launch(void* const* d_in, const int* in_sizes, int n_in_arrs,
                              void* d_out, int out_size, void* d_ws, size_t ws_size,
                              hipStream_t stream) {
    (void)n_in_arrs; (void)d_ws; (void)ws_size;
    const float* x       = (const float*)d_in[0];
    const int*   nn_idx  = (const int*)  d_in[1];   // int32 per harness convention
    const float* nn_dist = (const float*)d_in[2];
    float*       out     = (float*)d_out;

    constexpr int C = 64, K = 4;
    const int n_out = in_sizes[2] / K;                  // nn_dist is (N_out, K)
    const int B     = out_size / (n_out * C);           // out is (B, N_out, C)
    const int n_in  = in_sizes[0] / (B * C);            // x is (B, N_in, C)

    dim3 block(256);
    dim3 grid((n_out + 15) / 16);                       // 16 points per block

    if (B == 4) {
        iwd_gather_kernel<4><<<grid, block, 0, stream>>>(x, nn_idx, nn_dist, out,
                                                         n_in, n_out);
    } else {
        iwd_gather_kernel_rt<<<grid, block, 0, stream>>>(x, nn_idx, nn_dist, out,
                                                         n_in, n_out, B);
    }
}